// TAGClassifier_71176198029900
// MI455X (gfx1250) — compile-verified
//
#include <hip/hip_runtime.h>
#include <hip/hip_bf16.h>

#define NNODES 40000
#define NEDGES 640000
#define DIM    128
#define NCLS   10
#define NGRAPH 16

typedef float v2f __attribute__((ext_vector_type(2)));
typedef float v8f __attribute__((ext_vector_type(8)));

// ---------------------------------------------------------------- utilities

__global__ void zero_kernel(float* __restrict__ p, int n) {
    int i = blockIdx.x * blockDim.x + threadIdx.x;
    if (i < n) p[i] = 0.0f;
}

__global__ void deg_kernel(const int* __restrict__ dst, float* __restrict__ deg, int nE) {
    int e = blockIdx.x * blockDim.x + threadIdx.x;
    if (e < nE) atomicAdd(&deg[dst[e]], 1.0f);
}

__global__ void norm_kernel(float* __restrict__ norm, int n) {
    int i = blockIdx.x * blockDim.x + threadIdx.x;
    if (i < n) norm[i] = 1.0f / sqrtf(fmaxf(norm[i], 1.0f));
}

// dst[i] = src[i] * norm[node]; works in place (dst == src) as well
__global__ void scale_kernel(float* __restrict__ dst, const float* __restrict__ src,
                             const float* __restrict__ norm, int total) {
    int i = blockIdx.x * blockDim.x + threadIdx.x;
    if (i < total) dst[i] = src[i] * norm[i >> 7];
}

// acc[dst[e]][:] += xs[src[e]][:]  -- one wave per edge, float4 per lane
__global__ void edge_scatter_kernel(const int* __restrict__ src, const int* __restrict__ dst,
                                    const float* __restrict__ xs, float* __restrict__ acc,
                                    int nE) {
    int tid = blockIdx.x * blockDim.x + threadIdx.x;
    int e = tid >> 5;
    if (e >= nE) return;
    int c4 = (tid & 31) << 2;                    // feature chunk 0..124
    int s = src[e], d = dst[e];
    const float4 v = *(const float4*)(xs + (size_t)s * DIM + c4);
    float* p = acc + (size_t)d * DIM + c4;
    atomicAdd(p + 0, v.x);
    atomicAdd(p + 1, v.y);
    atomicAdd(p + 2, v.z);
    atomicAdd(p + 3, v.w);
}

// ------------------------------------------------ WMMA fp32 GEMM (K = 3*128)
// out[M x 128] = relu([A0|A1|A2][M x 384] @ W[384 x 128] + bias)
// One wave computes a 16-row x 128-col strip: 8 accumulator tiles,
// 96 k-steps of V_WMMA_F32_16X16X4_F32 each stepping 8 tiles.
__global__ void __launch_bounds__(256)
gemm_tag_kernel(const float* __restrict__ A0, const float* __restrict__ A1,
                const float* __restrict__ A2, const float* __restrict__ W,
                const float* __restrict__ bias, float* __restrict__ out,
                int mtiles, int do_relu) {
    int wave = (blockIdx.x * blockDim.x + threadIdx.x) >> 5;
    if (wave >= mtiles) return;                  // wave-uniform: EXEC stays all-1s
    int lane = threadIdx.x & 31;
    int half = lane >> 4;                        // 0: K%4 in {0,1} / M 0-7 of C
    int l    = lane & 15;
    int row  = wave * 16 + l;                    // A-matrix row for this lane

    v8f c[8];
#pragma unroll
    for (int i = 0; i < 8; ++i)
        c[i] = (v8f){0.f, 0.f, 0.f, 0.f, 0.f, 0.f, 0.f, 0.f};

    const float* segs[3] = {A0, A1, A2};
#pragma unroll 1
    for (int seg = 0; seg < 3; ++seg) {
        const float* arow = segs[seg] + (size_t)row * DIM + half * 2;
        const float* wseg = W + (size_t)seg * DIM * DIM + (size_t)(half * 2) * DIM + l;
#pragma unroll 1
        for (int kb = 0; kb < DIM; kb += 4) {
            // A 16x4: lanes 0-15 hold K=kb+0,kb+1 ; lanes 16-31 hold K=kb+2,kb+3
            v2f a = *(const v2f*)(arow + kb);
            const float* wr = wseg + (size_t)kb * DIM;
#pragma unroll
            for (int nt = 0; nt < 8; ++nt) {
                // B 4x16: v0 = rows K+0 (lanes 0-15) / K+2 (lanes 16-31), v1 = K+1/K+3
                v2f b;
                b.x = wr[nt * 16];
                b.y = wr[nt * 16 + DIM];
                c[nt] = __builtin_amdgcn_wmma_f32_16x16x4_f32(
                    false, a, false, b, (short)0, c[nt], false, false);
            }
        }
    }

    // C tile layout: VGPR i -> row (half?8:0)+i, col l (within 16-wide tile)
    int rbase = wave * 16 + half * 8;
#pragma unroll
    for (int nt = 0; nt < 8; ++nt) {
        int col = nt * 16 + l;
        float bv = bias[col];
#pragma unroll
        for (int i = 0; i < 8; ++i) {
            float v = c[nt][i] + bv;
            if (do_relu) v = fmaxf(v, 0.0f);
            out[(size_t)(rbase + i) * DIM + col] = v;
        }
    }
}

// -------------------------------------------------------------- pool + head

// segment_max of non-negative (post-ReLU) values: int bit-pattern atomicMax
__global__ void pool_kernel(const float* __restrict__ x, const int* __restrict__ gid,
                            float* __restrict__ pooled, int total) {
    int i = blockIdx.x * blockDim.x + threadIdx.x;
    if (i >= total) return;
    int node = i >> 7, f = i & 127;
    int g = gid[node];
    atomicMax((int*)&pooled[g * DIM + f], __float_as_int(x[i]));
}

__global__ void classify_kernel(const float* __restrict__ pooled, const float* __restrict__ Wc,
                                const float* __restrict__ bc, float* __restrict__ out) {
    int t = threadIdx.x;
    if (t >= NGRAPH * NCLS) return;
    int g = t / NCLS, cc = t % NCLS;
    float acc = bc[cc];
    for (int k = 0; k < DIM; ++k) acc += pooled[g * DIM + k] * Wc[k * NCLS + cc];
    out[t] = acc;
}

// ------------------------------------------------------------------- driver

extern "C" void kernel_launch(void* const* d_in, const int* in_sizes, int n_in,
                              void* d_out, int out_size, void* d_ws, size_t ws_size,
                              hipStream_t stream) {
    const float* h   = (const float*)d_in[0];
    const int*   src = (const int*)d_in[1];
    const int*   dst = (const int*)d_in[2];
    const int*   gid = (const int*)d_in[3];
    const float* W1  = (const float*)d_in[4];
    const float* b1  = (const float*)d_in[5];
    const float* W2  = (const float*)d_in[6];
    const float* b2  = (const float*)d_in[7];
    const float* Wc  = (const float*)d_in[8];
    const float* bc  = (const float*)d_in[9];
    float* out = (float*)d_out;

    const size_t NF = (size_t)NNODES * DIM;      // 5,120,000 floats
    float* ws     = (float*)d_ws;
    float* norm   = ws;                          // 40960
    float* bufA   = norm + 40960;                // propagation hop 1
    float* bufB   = bufA + NF;                   // propagation hop 2
    float* bufC   = bufB + NF;                   // scaled-x scratch / layer-2 output
    float* h1     = bufC + NF;                   // layer-1 output
    float* pooled = h1 + NF;                     // 2048

    const int T = 256;
    const int gN   = (NNODES + T - 1) / T;
    const int gNF  = (int)((NF + T - 1) / T);
    const int gE   = (NEDGES + T - 1) / T;
    const int gE32 = (NEDGES * 32 + T - 1) / T;
    const int mtiles = NNODES / 16;              // 2500
    const int gGemm  = (mtiles + 7) / 8;

    // degree -> norm = max(deg,1)^(-1/2)
    zero_kernel<<<gN, T, 0, stream>>>(norm, NNODES);
    deg_kernel<<<gE, T, 0, stream>>>(dst, norm, NEDGES);
    norm_kernel<<<gN, T, 0, stream>>>(norm, NNODES);

    // ---- TAGConv layer 1 (x = h) ----
    scale_kernel<<<gNF, T, 0, stream>>>(bufC, h, norm, (int)NF);
    zero_kernel<<<gNF, T, 0, stream>>>(bufA, (int)NF);
    edge_scatter_kernel<<<gE32, T, 0, stream>>>(src, dst, bufC, bufA, NEDGES);
    scale_kernel<<<gNF, T, 0, stream>>>(bufA, bufA, norm, (int)NF);      // hop 1

    scale_kernel<<<gNF, T, 0, stream>>>(bufC, bufA, norm, (int)NF);
    zero_kernel<<<gNF, T, 0, stream>>>(bufB, (int)NF);
    edge_scatter_kernel<<<gE32, T, 0, stream>>>(src, dst, bufC, bufB, NEDGES);
    scale_kernel<<<gNF, T, 0, stream>>>(bufB, bufB, norm, (int)NF);      // hop 2

    gemm_tag_kernel<<<gGemm, T, 0, stream>>>(h, bufA, bufB, W1, b1, h1, mtiles, 1);

    // ---- TAGConv layer 2 (x = h1) ----
    scale_kernel<<<gNF, T, 0, stream>>>(bufC, h1, norm, (int)NF);
    zero_kernel<<<gNF, T, 0, stream>>>(bufA, (int)NF);
    edge_scatter_kernel<<<gE32, T, 0, stream>>>(src, dst, bufC, bufA, NEDGES);
    scale_kernel<<<gNF, T, 0, stream>>>(bufA, bufA, norm, (int)NF);

    scale_kernel<<<gNF, T, 0, stream>>>(bufC, bufA, norm, (int)NF);
    zero_kernel<<<gNF, T, 0, stream>>>(bufB, (int)NF);
    edge_scatter_kernel<<<gE32, T, 0, stream>>>(src, dst, bufC, bufB, NEDGES);
    scale_kernel<<<gNF, T, 0, stream>>>(bufB, bufB, norm, (int)NF);

    gemm_tag_kernel<<<gGemm, T, 0, stream>>>(h1, bufA, bufB, W2, b2, bufC, mtiles, 1);

    // ---- graph max-pool + linear head ----
    zero_kernel<<<(NGRAPH * DIM + T - 1) / T, T, 0, stream>>>(pooled, NGRAPH * DIM);
    pool_kernel<<<gNF, T, 0, stream>>>(bufC, gid, pooled, (int)NF);
    classify_kernel<<<1, 192, 0, stream>>>(pooled, Wc, bc, out);
}